// DecoderRNN_30331059045059
// MI455X (gfx1250) — compile-verified
//
#include <hip/hip_runtime.h>
#include <hip/hip_bf16.h>

// Problem dims (fixed by the reference)
#define B_   128
#define T_   20
#define V_   32000
#define FEMB 2048
#define EMB_ 25
#define D_   25
#define D2_  625
#define IN_  675   // EMB + D2 + D
#define EPS_ 1e-5f

typedef float v2f __attribute__((ext_vector_type(2)));
typedef float v8f __attribute__((ext_vector_type(8)));

__device__ __forceinline__ float sigmoidf_(float v) { return 1.0f / (1.0f + __expf(-v)); }

// ---------------------------------------------------------------------------
// BatchNorm1d (training-mode batch stats), per-feature-column thread.
// ---------------------------------------------------------------------------
__global__ void bn_kernel(const float* __restrict__ feat,
                          const float* __restrict__ gamma,
                          const float* __restrict__ beta,
                          float* __restrict__ fn) {
  int f = blockIdx.x * blockDim.x + threadIdx.x;
  if (f >= FEMB) return;
  float s = 0.0f;
  for (int b = 0; b < B_; ++b) s += feat[b * FEMB + f];
  float mu = s * (1.0f / B_);
  float v = 0.0f;
  for (int b = 0; b < B_; ++b) {
    float d = feat[b * FEMB + f] - mu;
    v += d * d;
  }
  v *= (1.0f / B_);
  float inv = rsqrtf(v + EPS_);
  float g = gamma[f], be = beta[f];
  for (int b = 0; b < B_; ++b)
    fn[b * FEMB + f] = (feat[b * FEMB + f] - mu) * inv * g + be;
}

__global__ void zero_kernel(float* __restrict__ p, int n) {
  int i = blockIdx.x * blockDim.x + threadIdx.x;
  if (i < n) p[i] = 0.0f;
}

// ---------------------------------------------------------------------------
// Generic fp32 WMMA GEMM: C[m,n] = sum_k A[m,k]*Bt[n,k] (+ optional 2nd pass
// A2/Bt2 with K2) + bias1[n] + bias2[n].
//
// Block tile 64(M) x 128(N), 256 threads = 8 wave32 waves, each wave owns a
// 32x32 region = 2x2 fragments of V_WMMA_F32_16X16X4_F32.
// K staged via double-buffered LDS in chunks of 16 (register-staged global
// loads overlap the WMMA work; one barrier per chunk).
// Row/col indices are CLAMPED once up front: out-of-range rows/cols only feed
// fragments whose results are discarded by the guarded epilogue, so no
// per-chunk bounds checks are needed. Only the K-tail chunk zero-fills.
// Requirement: M,N >= 1; grid covers ceil(M/64) x ceil(N/128).
// ---------------------------------------------------------------------------
#define TM 64
#define TN 128
#define TK 16
#define AS_LD 17           // TK + 1  (conflict-free strided reads)
#define BS_LD 132          // TN + 4
#define ASZ (TM * AS_LD)   // 1088 floats
#define BSZ (TK * BS_LD)   // 2112 floats

__global__ __launch_bounds__(256)
void wmma_gemm_kernel(const float* __restrict__ A,  int lda,
                      const float* __restrict__ Bt, int ldb, int K1,
                      const float* __restrict__ A2, int lda2,
                      const float* __restrict__ Bt2, int ldb2, int K2,
                      const float* __restrict__ bias1,
                      const float* __restrict__ bias2,
                      float* __restrict__ C, int ldc, int M, int N) {
  __shared__ float As[2 * ASZ];
  __shared__ float Bs[2 * BSZ];

  const int tid  = threadIdx.x;
  const int lane = tid & 31;
  const int wid  = tid >> 5;      // 0..7
  const int wm   = wid & 1;       // wave M index (2)
  const int wn   = wid >> 1;      // wave N index (4)
  const int half = lane >> 4;     // 0/1 (lane group)
  const int l15  = lane & 15;
  const int row0 = blockIdx.y * TM;
  const int col0 = blockIdx.x * TN;

  // staging thread mapping: k = tid&15 (contiguous K -> coalesced), 16 rows/cols group
  const int sk = tid & 15;
  const int rb = tid >> 4;        // 0..15

  v8f c[2][2];
#pragma unroll
  for (int i = 0; i < 2; ++i)
#pragma unroll
    for (int j = 0; j < 2; ++j)
#pragma unroll
      for (int e = 0; e < 8; ++e) c[i][j][e] = 0.0f;

  const float* Ap[2] = {A, A2};
  const float* Bp[2] = {Bt, Bt2};
  const int ldap[2]  = {lda, lda2};
  const int ldbp[2]  = {ldb, ldb2};
  const int Kp[2]    = {K1, K2};

  // clamped row/col indices (loop-invariant across both passes)
  int arow[4], bcol[8];
#pragma unroll
  for (int i = 0; i < 4; ++i) {
    int gm = row0 + rb + i * 16;
    arow[i] = (gm < M) ? gm : (M - 1);
  }
#pragma unroll
  for (int i = 0; i < 8; ++i) {
    int gn = col0 + rb + i * 16;
    bcol[i] = (gn < N) ? gn : (N - 1);
  }

  for (int pass = 0; pass < 2; ++pass) {
    const float* __restrict__ Ax = Ap[pass];
    if (Ax == nullptr) continue;
    const float* __restrict__ Bx = Bp[pass];
    const int ldA = ldap[pass], ldB = ldbp[pass], K = Kp[pass];
    const int nch = (K + TK - 1) / TK;

    // per-thread 32-bit element offsets (add k0 each chunk)
    int aoff[4], boff[8];
#pragma unroll
    for (int i = 0; i < 4; ++i) aoff[i] = arow[i] * ldA + sk;
#pragma unroll
    for (int i = 0; i < 8; ++i) boff[i] = bcol[i] * ldB + sk;

    float rA[4], rB[8];

    // ---- loadRegs(chunk) ----
    auto loadRegs = [&](int ch) {
      const int kbase = ch * TK;
      if (kbase + TK <= K) {            // uniform (scalar) branch
#pragma unroll
        for (int i = 0; i < 4; ++i) rA[i] = Ax[aoff[i] + kbase];
#pragma unroll
        for (int i = 0; i < 8; ++i) rB[i] = Bx[boff[i] + kbase];
      } else {                          // K-tail: zero-fill beyond K
        const bool ok = (kbase + sk) < K;
#pragma unroll
        for (int i = 0; i < 4; ++i) rA[i] = ok ? Ax[aoff[i] + kbase] : 0.0f;
#pragma unroll
        for (int i = 0; i < 8; ++i) rB[i] = ok ? Bx[boff[i] + kbase] : 0.0f;
      }
    };
    // ---- storeRegs(buf) ----
    auto storeRegs = [&](int buf) {
      float* a = As + buf * ASZ;
      float* b = Bs + buf * BSZ;
#pragma unroll
      for (int i = 0; i < 4; ++i) a[(rb + i * 16) * AS_LD + sk] = rA[i];
#pragma unroll
      for (int i = 0; i < 8; ++i) b[sk * BS_LD + rb + i * 16] = rB[i];
    };
    // ---- compute(buf) ----
    auto compute = [&](int buf) {
      const float* a = As + buf * ASZ;
      const float* b = Bs + buf * BSZ;
#pragma unroll
      for (int kk = 0; kk < 4; ++kk) {
        const int kb = kk * 4 + half * 2;  // this lane-group's K pair
        v2f afr[2], bfr[2];
#pragma unroll
        for (int i = 0; i < 2; ++i) {
          int m = wm * 32 + i * 16 + l15;
          afr[i].x = a[m * AS_LD + kb];
          afr[i].y = a[m * AS_LD + kb + 1];
          int n = wn * 32 + i * 16 + l15;
          bfr[i].x = b[kb * BS_LD + n];
          bfr[i].y = b[(kb + 1) * BS_LD + n];
        }
#pragma unroll
        for (int i = 0; i < 2; ++i)
#pragma unroll
          for (int j = 0; j < 2; ++j)
            c[i][j] = __builtin_amdgcn_wmma_f32_16x16x4_f32(
                false, afr[i], false, bfr[j], (short)0, c[i][j], false, false);
      }
    };

    // ---- software pipeline: 1 barrier per chunk ----
    loadRegs(0);
    storeRegs(0);
    __syncthreads();
    for (int ch = 0; ch < nch; ++ch) {
      const bool more = (ch + 1 < nch);
      if (more) loadRegs(ch + 1);       // global -> regs, overlapped with WMMA
      compute(ch & 1);
      if (more) storeRegs((ch + 1) & 1);
      __syncthreads();
    }
  }

  // --- epilogue: C-fragment layout: VGPR r -> row (half*8 + r), col l15 ---
#pragma unroll
  for (int j = 0; j < 2; ++j) {
    int gn = col0 + wn * 32 + j * 16 + l15;
    float bv = 0.0f;
    if (gn < N) {
      if (bias1) bv += bias1[gn];
      if (bias2) bv += bias2[gn];
    }
#pragma unroll
    for (int i = 0; i < 2; ++i)
#pragma unroll
      for (int r = 0; r < 8; ++r) {
        int gm = row0 + wm * 32 + i * 16 + half * 8 + r;
        if (gm < M && gn < N)
          C[(size_t)gm * ldc + gn] = c[i][j][r] + bv;
      }
  }
}

// ---------------------------------------------------------------------------
// Assemble x = [cU | word | cS] per step (with embedding gather).
// ---------------------------------------------------------------------------
__global__ void assemble_x_kernel(float* __restrict__ x,
                                  const float* __restrict__ cU,
                                  const float* __restrict__ cS,
                                  const float* __restrict__ embW,
                                  const int* __restrict__ captions,
                                  int step) {
  int idx = blockIdx.x * blockDim.x + threadIdx.x;
  if (idx >= B_ * IN_) return;
  int b = idx / IN_;
  int k = idx - b * IN_;
  float v;
  if (k < D_) {
    v = cU[b * D_ + k];
  } else if (k < D_ + EMB_) {
    int w = (step == 0) ? 0 : captions[b * T_ + step];
    v = embW[w * EMB_ + (k - D_)];
  } else {
    v = cS[b * D2_ + (k - D_ - EMB_)];
  }
  x[idx] = v;
}

// ---------------------------------------------------------------------------
// Per-sample tail of a step: LSTM-U gates (scalar, tiny), both cell updates,
// u = (tanh)(hU @ wu_W^T + wu_b), ft = u_mat @ hS_mat  -> word_out slice.
// One block per batch sample; all cross-phase deps inside the block.
// ---------------------------------------------------------------------------
__global__ __launch_bounds__(256)
void step_tail_kernel(const float* __restrict__ gatesS,  // [B, 4*D2]
                      const float* __restrict__ x,       // [B, IN]
                      const float* __restrict__ WihU,    // [4D, IN]
                      const float* __restrict__ WhhU,    // [4D, D]
                      const float* __restrict__ bihU,
                      const float* __restrict__ bhhU,
                      float* __restrict__ hS, float* __restrict__ cS,
                      float* __restrict__ hU, float* __restrict__ cU,
                      const float* __restrict__ wuW,     // [D2, D]
                      const float* __restrict__ wub,     // [D2]
                      float* __restrict__ wout,          // word_out[t+1]: [B, D2]
                      int step0) {
  __shared__ float xs[IN_];
  __shared__ float hUold[D_];
  __shared__ float gU[4 * D_];
  __shared__ float hUs[D_];
  __shared__ float hSs[D2_];
  __shared__ float us[D2_];

  const int b   = blockIdx.x;
  const int tid = threadIdx.x;

  // Phase 0: stage x[b,:] and old hU[b,:] into LDS
  for (int k = tid; k < IN_; k += 256) xs[k] = x[b * IN_ + k];
  if (tid < D_) hUold[tid] = hU[b * D_ + tid];
  __syncthreads();

  // Phase 1: LSTM-U gates (100 outputs, K = 675 + 25)
  if (tid < 4 * D_) {
    float acc = bihU[tid] + bhhU[tid];
    const float* wr = WihU + tid * IN_;
    for (int k = 0; k < IN_; ++k) acc += xs[k] * wr[k];
    const float* w2 = WhhU + tid * D_;
#pragma unroll
    for (int k = 0; k < D_; ++k) acc += hUold[k] * w2[k];
    gU[tid] = acc;
  }
  __syncthreads();

  // Phase 2: cell updates (gate order i, f, g, o)
  if (tid < D_) {
    float gi = sigmoidf_(gU[tid]);
    float gf = sigmoidf_(gU[D_ + tid]);
    float gg = tanhf(gU[2 * D_ + tid]);
    float go = sigmoidf_(gU[3 * D_ + tid]);
    float cc = gf * cU[b * D_ + tid] + gi * gg;
    float hh = go * tanhf(cc);
    cU[b * D_ + tid] = cc;
    hU[b * D_ + tid] = hh;
    hUs[tid] = hh;
  }
  const float* gr = gatesS + (size_t)b * 4 * D2_;
  for (int n = tid; n < D2_; n += 256) {
    float gi = sigmoidf_(gr[n]);
    float gf = sigmoidf_(gr[D2_ + n]);
    float gg = tanhf(gr[2 * D2_ + n]);
    float go = sigmoidf_(gr[3 * D2_ + n]);
    float cc = gf * cS[b * D2_ + n] + gi * gg;
    float hh = go * tanhf(cc);
    cS[b * D2_ + n] = cc;
    hS[b * D2_ + n] = hh;
    hSs[n] = hh;
  }
  __syncthreads();

  // Phase 3: u = (tanh)(hU @ wu_W^T + wu_b)   (no tanh at step 0, faithful)
  for (int n = tid; n < D2_; n += 256) {
    float acc = wub[n];
    const float* wr = wuW + n * D_;
#pragma unroll
    for (int j = 0; j < D_; ++j) acc += hUs[j] * wr[j];
    us[n] = step0 ? acc : tanhf(acc);
  }
  __syncthreads();

  // Phase 4: ft[i,k] = sum_j u[i,j] * hS_mat[j,k]
  for (int idx = tid; idx < D2_; idx += 256) {
    int i = idx / D_, k = idx - i * D_;
    float acc = 0.0f;
#pragma unroll
    for (int j = 0; j < D_; ++j) acc += us[i * D_ + j] * hSs[j * D_ + k];
    wout[(size_t)b * D2_ + idx] = acc;
  }
}

// ---------------------------------------------------------------------------
extern "C" void kernel_launch(void* const* d_in, const int* in_sizes, int n_in,
                              void* d_out, int out_size, void* d_ws, size_t ws_size,
                              hipStream_t stream) {
  (void)in_sizes; (void)n_in; (void)out_size; (void)ws_size;

  const float* features = (const float*)d_in[0];
  const int*   captions = (const int*)  d_in[1];
  const float* embW     = (const float*)d_in[2];
  const float* WihS     = (const float*)d_in[3];
  const float* WhhS     = (const float*)d_in[4];
  const float* bihS     = (const float*)d_in[5];
  const float* bhhS     = (const float*)d_in[6];
  const float* WihU     = (const float*)d_in[7];
  const float* WhhU     = (const float*)d_in[8];
  const float* bihU     = (const float*)d_in[9];
  const float* bhhU     = (const float*)d_in[10];
  const float* fcW      = (const float*)d_in[11];
  const float* fcb      = (const float*)d_in[12];
  const float* szW      = (const float*)d_in[13];
  const float* szb      = (const float*)d_in[14];
  const float* wuW      = (const float*)d_in[15];
  const float* wub      = (const float*)d_in[16];
  const float* gamma    = (const float*)d_in[17];
  const float* beta     = (const float*)d_in[18];
  float* out = (float*)d_out;

  // Workspace layout (floats); total ~9.8 MB.
  float* ws     = (float*)d_ws;
  float* fn     = ws;  ws += B_ * FEMB;        // 262144
  float* x      = ws;  ws += B_ * IN_;         // 86400
  float* gatesS = ws;  ws += B_ * 4 * D2_;     // 320000
  float* hS     = ws;  ws += B_ * D2_;
  float* cS     = ws;  ws += B_ * D2_;
  float* hU     = ws;  ws += B_ * D_;
  float* cU     = ws;  ws += B_ * D_;
  float* wordo  = ws;  ws += (size_t)T_ * B_ * D2_;  // 1600000

  // 1. BatchNorm
  bn_kernel<<<(FEMB + 255) / 256, 256, 0, stream>>>(features, gamma, beta, fn);

  // 2. Init state + word_out[0] = 0
  zero_kernel<<<(B_ * D2_ + 255) / 256, 256, 0, stream>>>(hS, B_ * D2_);
  zero_kernel<<<(B_ * D_ + 255) / 256, 256, 0, stream>>>(hU, B_ * D_);
  zero_kernel<<<(B_ * D_ + 255) / 256, 256, 0, stream>>>(cU, B_ * D_);
  zero_kernel<<<(B_ * D2_ + 255) / 256, 256, 0, stream>>>(wordo, B_ * D2_);

  // 3. cS0 = fn @ szW^T + szb   (M=128, N=625, K=2048)
  {
    dim3 grid((D2_ + TN - 1) / TN, (B_ + TM - 1) / TM);
    wmma_gemm_kernel<<<grid, 256, 0, stream>>>(
        fn, FEMB, szW, FEMB, FEMB,
        nullptr, 0, nullptr, 0, 0,
        szb, nullptr, cS, D2_, B_, D2_);
  }

  // 4. 19 recurrent steps
  for (int t = 0; t < T_ - 1; ++t) {
    assemble_x_kernel<<<(B_ * IN_ + 255) / 256, 256, 0, stream>>>(
        x, cU, cS, embW, captions, t);

    // gatesS = x @ WihS^T + hS @ WhhS^T + bihS + bhhS  (fused two-pass WMMA GEMM)
    dim3 grid((4 * D2_ + TN - 1) / TN, (B_ + TM - 1) / TM);
    wmma_gemm_kernel<<<grid, 256, 0, stream>>>(
        x, IN_, WihS, IN_, IN_,
        hS, D2_, WhhS, D2_, D2_,
        bihS, bhhS, gatesS, 4 * D2_, B_, 4 * D2_);

    step_tail_kernel<<<B_, 256, 0, stream>>>(
        gatesS, x, WihU, WhhU, bihU, bhhU,
        hS, cS, hU, cU, wuW, wub,
        wordo + (size_t)(t + 1) * B_ * D2_, (t == 0) ? 1 : 0);
  }

  // 5. tag = word_out @ fcW^T + fcb  (M=2560, N=32000, K=625) -> full d_out
  {
    dim3 grid((V_ + TN - 1) / TN, (T_ * B_ + TM - 1) / TM);
    wmma_gemm_kernel<<<grid, 256, 0, stream>>>(
        wordo, D2_, fcW, D2_, D2_,
        nullptr, 0, nullptr, 0, 0,
        fcb, nullptr, out, V_, T_ * B_, V_);
  }
}